// HRF_1967095021987
// MI455X (gfx1250) — compile-verified
//
#include <hip/hip_runtime.h>

// HRF banded-Toeplitz materialization for MI455X (gfx1250).
// Pure store-bandwidth problem: 327.7 MB fp32 output, ~0 arithmetic intensity
// -> roofline floor ~14 us at 23.3 TB/s. No contraction exists, so no WMMA;
// the CDNA5-relevant path is wide non-temporal stores + ASYNCcnt LDS-DMA fill.
// - band chunks  : computed in registers, stored with non-temporal b128
// - zero chunks  : GLOBAL_STORE_ASYNC_FROM_LDS_B128 th:TH_STORE_NT (all lanes
//                  source the same 16 zero bytes in LDS)
// - taps         : 30-tap HRF kernel computed once per block into LDS

typedef float v4f __attribute__((ext_vector_type(4)));

#define T_LEN 400
#define L_LEN 30
#define ROW_CHUNKS 100              // 400 floats / 4 per row
#define CHUNKS_PER_MAT 40000        // 400 rows * 100 chunks
#define SPLITS 4
#define CHUNKS_PER_SPLIT (CHUNKS_PER_MAT / SPLITS)

__global__ __launch_bounds__(256)
void hrf_toeplitz_kernel(const float* __restrict__ theta, float* __restrict__ out) {
  __shared__ __align__(16) float zero16[4];   // async-store zero source
  __shared__ float kern[32];                  // 30 HRF taps (LDS broadcast)

  const int b     = blockIdx.x;
  const int split = blockIdx.y;
  const int tid   = threadIdx.x;

  if (tid < 4) zero16[tid] = 0.0f;

  if (tid < L_LEN) {
    // t = linspace(0, 30, 30000)[::1000] -> t[k] = k * (30000/29999); t[0] clipped to 1e-8
    const float th = theta[b];
    const float T_STEP = 30000.0f / 29999.0f;       // constant-folded reciprocal multiply
    float t = (tid == 0) ? 1e-8f : (float)tid * T_STEP;
    float t2  = t * t,   t4  = t2 * t2,  t5  = t4 * t;
    float t10 = t5 * t5, t15 = t10 * t5;
    float th2 = th * th, th4 = th2 * th2;
    float th6 = th4 * th2, th8 = th4 * th4, th16 = th8 * th8;
    float e = __expf(-th * t);
    const float INV_G6 = 1.0f / 120.0f;             // 1/Gamma(6)
    const float SCL_U  = 0.167f / 1.307674368e12f;  // C12/Gamma(16)
    // peak - 0.167*undershoot
    kern[tid] = th6 * t5 * e * INV_G6 - th16 * t15 * e * SCL_U;
  }
  __syncthreads();

  // low 32 bits of the flat address of a __shared__ object == wave-relative LDS byte address
  const unsigned lds_zero = (unsigned)(unsigned long long)(&zero16[0]);
  float* const base = out + (size_t)b * (size_t)(T_LEN * T_LEN);

  const int c0 = split * CHUNKS_PER_SPLIT;
  const int c1 = c0 + CHUNKS_PER_SPLIT;
  for (int c = c0 + tid; c < c1; c += 256) {
    float* gp = base + ((size_t)c << 2);        // chunk c -> 4 floats, 16B aligned
    const int j   = c / ROW_CHUNKS;             // output row (mul-hi by compiler)
    const int i0  = (c - j * ROW_CHUNKS) << 2;  // first column of chunk
    const int dhi = j - i0;                     // band index for column i0

    // chunk intersects the band iff any d in [dhi-3, dhi] lies in [0, 30)
    if (dhi >= 0 && (dhi - 3) < L_LEN) {
      const int d1 = dhi - 1, d2 = dhi - 2, d3 = dhi - 3;
      v4f v;
      v.x = (dhi < L_LEN)           ? kern[dhi] : 0.0f;  // dhi >= 0 guaranteed
      v.y = (d1 >= 0 && d1 < L_LEN) ? kern[d1]  : 0.0f;
      v.z = (d2 >= 0 && d2 < L_LEN) ? kern[d2]  : 0.0f;
      v.w = (d3 >= 0)               ? kern[d3]  : 0.0f;  // d3 < 30 guaranteed
      __builtin_nontemporal_store(v, (v4f*)gp);          // streaming b128 store
    } else {
      // pure-zero chunk: async DMA 16 zero bytes from LDS to global (ASYNCcnt path),
      // non-temporal so the 328 MB stream doesn't thrash the 192 MB L2
      asm volatile("global_store_async_from_lds_b128 %0, %1, off th:TH_STORE_NT"
                   :: "v"(gp), "v"(lds_zero)
                   : "memory");
    }
  }

  // drain async LDS->global stores (S_ENDPGM also implies wait-idle; explicit for clarity)
  asm volatile("s_wait_asynccnt 0x0" ::: "memory");
}

extern "C" void kernel_launch(void* const* d_in, const int* in_sizes, int n_in,
                              void* d_out, int out_size, void* d_ws, size_t ws_size,
                              hipStream_t stream) {
  const float* theta = (const float*)d_in[0];   // [B,1] fp32, B = 512
  float* out = (float*)d_out;                   // [B,400,400] fp32
  const int B = in_sizes[0];
  dim3 grid((unsigned)B, SPLITS, 1);
  hrf_toeplitz_kernel<<<grid, 256, 0, stream>>>(theta, out);
}